// GConvNetSimple_26310969655869
// MI455X (gfx1250) — compile-verified
//
#include <hip/hip_runtime.h>
#include <hip/hip_bf16.h>

typedef __attribute__((ext_vector_type(2))) float v2f;
typedef __attribute__((ext_vector_type(8))) float v8f;

// ---------------------------------------------------------------------------
// Utility: zero a u32/f32 region (grid-stride)
// ---------------------------------------------------------------------------
__global__ void zero_u32_kernel(unsigned int* __restrict__ p, size_t n) {
    size_t i = (size_t)blockIdx.x * blockDim.x + threadIdx.x;
    size_t stride = (size_t)gridDim.x * blockDim.x;
    for (; i < n; i += stride) p[i] = 0u;
}

// ---------------------------------------------------------------------------
// Degree counting: one thread per edge, u32 atomics
// ---------------------------------------------------------------------------
__global__ void degree_kernel(const int* __restrict__ src, const int* __restrict__ dst,
                              unsigned int* __restrict__ deg_out,
                              unsigned int* __restrict__ deg_in, int n_edges) {
    int e = blockIdx.x * blockDim.x + threadIdx.x;
    if (e >= n_edges) return;
    atomicAdd(&deg_out[src[e]], 1u);
    atomicAdd(&deg_in[dst[e]], 1u);
}

// norm = clip(deg,1)^-0.5
__global__ void norm_kernel(const unsigned int* __restrict__ deg_out,
                            const unsigned int* __restrict__ deg_in,
                            float* __restrict__ out_norm, float* __restrict__ in_norm,
                            int n_nodes) {
    int i = blockIdx.x * blockDim.x + threadIdx.x;
    if (i >= n_nodes) return;
    out_norm[i] = 1.0f / sqrtf(fmaxf((float)deg_out[i], 1.0f));
    in_norm[i]  = 1.0f / sqrtf(fmaxf((float)deg_in[i], 1.0f));
}

// ---------------------------------------------------------------------------
// Node projection with fp32 WMMA:  Y[16 nodes x KOUT] = (X * out_norm) @ W
// One wave handles a 16-node tile. V_WMMA_F32_16X16X4_F32, K stepped by 4.
//   A (16x4 f32, 2 VGPRs): lanes 0-15 hold K0/K1, lanes 16-31 K2/K3, M=lane&15
//   B (4x16 f32, 2 VGPRs): mirrored (lane&15 = N, lane half selects K pair)
//   C/D (16x16 f32, 8 VGPRs): VGPR r = row (half*8 + r), N = lane&15
// Branch-free operand setup:
//   - W staged once per block into zero-padded LDS tile -> unconditional ds_load
//   - X loads from clamped (always-valid) addresses, values masked by cndmask
// Store epilogue: wave-uniform full-tile fast path -> single lane predicate +
//   8-store clause at immediate offsets; guarded path only for the last tile.
// ---------------------------------------------------------------------------
template <int KIN, int KOUT>
__global__ void project_wmma_kernel(const float* __restrict__ X,
                                    const float* __restrict__ W,
                                    const float* __restrict__ nrm,
                                    float* __restrict__ Y, int n_nodes) {
    constexpr int PK = (KIN + 3) & ~3;          // K padded to multiple of 4
    __shared__ float sW[PK][16];                // zero-padded weight tile
    for (int t = threadIdx.x; t < PK * 16; t += blockDim.x) {
        const int kk = t >> 4;
        const int nn = t & 15;
        sW[kk][nn] = (kk < KIN && nn < KOUT) ? W[kk * KOUT + nn] : 0.0f;
    }
    __syncthreads();

    const int lane = threadIdx.x & 31;
    const int wave = (int)((blockIdx.x * blockDim.x + threadIdx.x) >> 5);
    const int node_base = wave * 16;
    if (node_base >= n_nodes) return;           // wave-uniform: EXEC all-ones

    const int m    = lane & 15;                 // A row / B,D column
    const int half = lane >> 4;                 // 0 or 1
    const int node = node_base + m;
    const bool nodeok = node < n_nodes;
    const int  nodec  = nodeok ? node : (n_nodes - 1);   // clamped, always valid
    const float scale = nodeok ? nrm[nodec] : 0.0f;
    const float* __restrict__ xrow = X + (size_t)nodec * KIN;

    v8f c = {};
    #pragma unroll
    for (int k0 = 0; k0 < KIN; k0 += 4) {
        const int ka = k0 + 2 * half;
        const int kb = ka + 1;
        // unconditional loads from clamped addresses; mask values (v_cndmask)
        const float ax = xrow[ka < KIN ? ka : 0];
        const float ay = xrow[kb < KIN ? kb : 0];
        v2f a, b;
        a.x = (ka < KIN) ? ax * scale : 0.0f;
        a.y = (kb < KIN) ? ay * scale : 0.0f;
        b.x = sW[ka][m];                        // padded tile: always in range
        b.y = sW[kb][m];
        c = __builtin_amdgcn_wmma_f32_16x16x4_f32(false, a, false, b,
                                                  (short)0, c, false, false);
    }

    const int nd0 = node_base + half * 8;       // first row this lane-half owns
    if (node_base + 16 <= n_nodes) {
        // full tile (wave-uniform): one lane predicate, immediate-offset stores
        if (m < KOUT) {
            float* __restrict__ yp = Y + (size_t)nd0 * KOUT + m;
            #pragma unroll
            for (int r = 0; r < 8; ++r) yp[r * KOUT] = c[r];
        }
    } else {
        // partial last tile: per-row guards (at most one wave in the grid)
        if (m < KOUT) {
            #pragma unroll
            for (int r = 0; r < 8; ++r) {
                const int nd = nd0 + r;
                if (nd < n_nodes) Y[(size_t)nd * KOUT + m] = c[r];
            }
        }
    }
}

// ---------------------------------------------------------------------------
// Edge scatter: AGG[dst] += Y[src], K floats per edge.
// Rows are 16B-aligned by workspace construction -> float4 (b128) loads.
// ---------------------------------------------------------------------------
template <int K>
__global__ void edge_agg_kernel(const int* __restrict__ src, const int* __restrict__ dst,
                                const float* __restrict__ Y, float* __restrict__ AGG,
                                int n_edges) {
    int e = blockIdx.x * blockDim.x + threadIdx.x;
    if (e >= n_edges) return;
    const int s = src[e];
    const int d = dst[e];
    const float4* __restrict__ yp = (const float4*)(Y + (size_t)s * K);
    float* ap = AGG + (size_t)d * K;
    #pragma unroll
    for (int q = 0; q < K / 4; ++q) {
        const float4 v = yp[q];
        atomicAdd(ap + 4 * q + 0, v.x);
        atomicAdd(ap + 4 * q + 1, v.y);
        atomicAdd(ap + 4 * q + 2, v.z);
        atomicAdd(ap + 4 * q + 3, v.w);
    }
}

// ---------------------------------------------------------------------------
// H = relu(AGG * in_norm + bias)
// ---------------------------------------------------------------------------
template <int K>
__global__ void relu_bias_kernel(const float* __restrict__ AGG,
                                 const float* __restrict__ in_norm,
                                 const float* __restrict__ bias,
                                 float* __restrict__ H, int n_nodes) {
    int i = blockIdx.x * blockDim.x + threadIdx.x;
    if (i >= n_nodes) return;
    const float s = in_norm[i];
    #pragma unroll
    for (int j = 0; j < K; ++j) {
        float v = AGG[(size_t)i * K + j] * s + bias[j];
        H[(size_t)i * K + j] = fmaxf(v, 0.0f);
    }
}

// ---------------------------------------------------------------------------
// Fused layer-2 epilogue + graph pooling:
//   h2 = relu(agg2*in_norm + b2); gsum[gid] += h2; gcnt[gid] += 1
// ---------------------------------------------------------------------------
__global__ void pool_kernel(const float* __restrict__ AGG2,
                            const float* __restrict__ in_norm,
                            const float* __restrict__ b2,
                            const int* __restrict__ gids,
                            float* __restrict__ gsum, float* __restrict__ gcnt,
                            int n_nodes) {
    int i = blockIdx.x * blockDim.x + threadIdx.x;
    if (i >= n_nodes) return;
    const float s = in_norm[i];
    const int g = gids[i];
    #pragma unroll
    for (int j = 0; j < 4; ++j) {
        float v = fmaxf(AGG2[(size_t)i * 4 + j] * s + b2[j], 0.0f);
        atomicAdd(&gsum[g * 4 + j], v);
    }
    atomicAdd(&gcnt[g], 1.0f);
}

// out[g] = sigmoid(mean @ Wo + bo)
__global__ void out_kernel(const float* __restrict__ gsum, const float* __restrict__ gcnt,
                           const float* __restrict__ Wo, const float* __restrict__ bo,
                           float* __restrict__ out, int n_graphs) {
    int g = blockIdx.x * blockDim.x + threadIdx.x;
    if (g >= n_graphs) return;
    const float c = fmaxf(gcnt[g], 1.0f);
    float acc = bo[0];
    #pragma unroll
    for (int j = 0; j < 4; ++j) acc += (gsum[g * 4 + j] / c) * Wo[j];
    out[g] = 1.0f / (1.0f + expf(-acc));
}

// ---------------------------------------------------------------------------
extern "C" void kernel_launch(void* const* d_in, const int* in_sizes, int n_in,
                              void* d_out, int out_size, void* d_ws, size_t ws_size,
                              hipStream_t stream) {
    const float* features = (const float*)d_in[0];
    const int*   src      = (const int*)d_in[1];
    const int*   dst      = (const int*)d_in[2];
    const int*   gids     = (const int*)d_in[3];
    const float* W1 = (const float*)d_in[4];
    const float* b1 = (const float*)d_in[5];
    const float* W2 = (const float*)d_in[6];
    const float* b2 = (const float*)d_in[7];
    const float* Wo = (const float*)d_in[8];
    const float* bo = (const float*)d_in[9];
    float* out = (float*)d_out;

    const int n_edges  = in_sizes[1];
    const int n_nodes  = in_sizes[3];
    const int n_graphs = out_size;
    const size_t n = (size_t)n_nodes;

    // Workspace layout (bytes, relative to d_ws):
    //   [0      :  4n) deg_out u32
    //   [4n     :  8n) deg_in  u32
    //   [8n     : 12n) out_norm f32
    //   [12n    : 16n) in_norm  f32
    //   [16n    : 48n) y1 (n x 8)  -- reused as h1 after edge_agg1
    //   [48n    : 80n) agg1 (n x 8) -- lower half reused as y2, upper as agg2
    //   [80n    : +20*n_graphs) gsum (n_graphs x 4) then gcnt (n_graphs)
    char* ws = (char*)d_ws;
    unsigned int* deg_out = (unsigned int*)(ws + 0);
    unsigned int* deg_in  = (unsigned int*)(ws + 4 * n);
    float* out_norm = (float*)(ws + 8 * n);
    float* in_norm  = (float*)(ws + 12 * n);
    float* y1   = (float*)(ws + 16 * n);  // n x 8
    float* h1   = y1;                     // reuse after agg1 is built
    float* agg1 = (float*)(ws + 48 * n);  // n x 8
    float* y2   = (float*)(ws + 48 * n);  // n x 4 (agg1 lower half, agg1 dead)
    float* agg2 = (float*)(ws + 64 * n);  // n x 4 (agg1 upper half, agg1 dead)
    float* gsum = (float*)(ws + 80 * n);
    float* gcnt = (float*)(ws + 80 * n + 16 * (size_t)n_graphs);

    const int B = 256;
    const int gridE = (n_edges + B - 1) / B;
    const int gridN = (n_nodes + B - 1) / B;
    const int gridG = (n_graphs + B - 1) / B;
    // project: 16 nodes/wave, 8 waves/block -> 128 nodes per block
    const int gridP = (n_nodes + 128 - 1) / 128;

    // 1. degrees + norms
    zero_u32_kernel<<<4096, B, 0, stream>>>((unsigned int*)ws, 2 * n); // deg_out+deg_in
    degree_kernel<<<gridE, B, 0, stream>>>(src, dst, deg_out, deg_in, n_edges);
    norm_kernel<<<gridN, B, 0, stream>>>(deg_out, deg_in, out_norm, in_norm, n_nodes);

    // 2. layer 1: y1 = (x*out_norm)@W1 ; agg1 = scatter; h1 = relu(agg1*in_norm+b1)
    zero_u32_kernel<<<4096, B, 0, stream>>>((unsigned int*)agg1, 8 * n);
    project_wmma_kernel<10, 8><<<gridP, B, 0, stream>>>(features, W1, out_norm, y1, n_nodes);
    edge_agg_kernel<8><<<gridE, B, 0, stream>>>(src, dst, y1, agg1, n_edges);
    relu_bias_kernel<8><<<gridN, B, 0, stream>>>(agg1, in_norm, b1, h1, n_nodes);

    // 3. layer 2 (agg1 now dead; its space is reused for y2/agg2)
    zero_u32_kernel<<<4096, B, 0, stream>>>((unsigned int*)agg2, 4 * n);
    project_wmma_kernel<8, 4><<<gridP, B, 0, stream>>>(h1, W2, out_norm, y2, n_nodes);
    edge_agg_kernel<4><<<gridE, B, 0, stream>>>(src, dst, y2, agg2, n_edges);

    // 4. pooling + readout
    zero_u32_kernel<<<64, B, 0, stream>>>((unsigned int*)gsum, 5 * (size_t)n_graphs);
    pool_kernel<<<gridN, B, 0, stream>>>(agg2, in_norm, b2, gids, gsum, gcnt, n_nodes);
    out_kernel<<<gridG, B, 0, stream>>>(gsum, gcnt, Wo, bo, out, n_graphs);
}